// TransitionDown_66958540144766
// MI455X (gfx1250) — compile-verified
//
#include <hip/hip_runtime.h>
#include <stdint.h>

// Problem shape (fixed by the reference harness)
#define FPS_B     32
#define FPS_N     16384
#define FPS_NS    4096           // 0.25 * N
#define FPS_C     128
#define FPS_BLOCK 1024
#define FPS_P     (FPS_N / FPS_BLOCK)   // 16 points per thread

typedef __attribute__((ext_vector_type(4))) unsigned int u32x4;
typedef __attribute__((ext_vector_type(8))) int          i32x8;
typedef __attribute__((ext_vector_type(4))) int          i32x4;

// Dynamic LDS layout:
//   [0, N*3*4)        float sxyz[N*3]   (xyz slab staged by the TDM)
//   then              float redv[32], int redi[32], int swin
#define LDS_XYZ_BYTES (FPS_N * 3 * 4)
#define LDS_TOTAL     (LDS_XYZ_BYTES + 32 * 4 + 32 * 4 + 16)

__global__ __launch_bounds__(FPS_BLOCK) void fps_kernel(const float* __restrict__ xyz,
                                                        int* __restrict__ idx_out) {
  extern __shared__ float lds[];
  float* sxyz = lds;
  float* redv = lds + FPS_N * 3;
  int*   redi = (int*)(redv + 32);
  int*   swin = redi + 32;

  const int b    = blockIdx.x;
  const int tid  = threadIdx.x;
  const int lane = tid & 31;
  const int wave = tid >> 5;
  const float* gx = xyz + (size_t)b * FPS_N * 3;

  // ---- Stage this batch's xyz (192 KB) into LDS with the Tensor Data Mover ----
  if (tid < 32) {  // wave 0 only issues the TDM descriptor (EXEC is ignored by TDM)
    const uint64_t ga  = (uint64_t)(uintptr_t)gx;
    const uint32_t lof = (uint32_t)(uintptr_t)sxyz;  // LDS offset = addr[31:0]
    u32x4 g0;
    g0[0] = 1u;                                                 // count=1, user descriptor
    g0[1] = lof;                                                // lds_addr
    g0[2] = (uint32_t)ga;                                       // global_addr[31:0]
    g0[3] = ((uint32_t)(ga >> 32) & 0x01FFFFFFu) | (2u << 30);  // addr[56:32] | type=2
    i32x8 g1;
    g1[0] = (int)(2u << 16);                            // data_size = 4 bytes
    g1[1] = (int)((unsigned)(FPS_N * 3) << 16);         // tensor_dim0 lo16 (49152)
    g1[2] = (int)(1u << 16);                            // tensor_dim0 hi=0, tensor_dim1 = 1
    g1[3] = (int)((unsigned)(FPS_N * 3) << 16);         // tile_dim0 = 49152 (1-D tile)
    g1[4] = 0;                                          // tile_dim1/2 unused
    g1[5] = FPS_N * 3;                                  // tensor_dim0_stride
    g1[6] = 0;
    g1[7] = 0;
    i32x4 gz4 = {0, 0, 0, 0};
    i32x8 gz8 = {0, 0, 0, 0, 0, 0, 0, 0};
    __builtin_amdgcn_tensor_load_to_lds(g0, g1, gz4, gz4, gz8, 0);
    __builtin_amdgcn_s_wait_tensorcnt(0);
  }
  __syncthreads();

  // ---- Pull this thread's 16 points into registers (one-time LDS read) ----
  float px[FPS_P], py[FPS_P], pz[FPS_P], dist[FPS_P];
#pragma unroll
  for (int k = 0; k < FPS_P; ++k) {
    const int p = tid + k * FPS_BLOCK;   // (3*tid) mod 64 banks -> conflict-free
    px[k] = sxyz[3 * p + 0];
    py[k] = sxyz[3 * p + 1];
    pz[k] = sxyz[3 * p + 2];
    dist[k] = 3.402823466e+38f;
  }

  int cur = 0;  // deterministic start at index 0
  for (int it = 0; it < FPS_NS; ++it) {
    // Broadcast centroid coords (same-address LDS read = broadcast)
    const float cx = sxyz[3 * cur + 0];
    const float cy = sxyz[3 * cur + 1];
    const float cz = sxyz[3 * cur + 2];

    float bestv = -1.0f;
    int   besti = 0x7fffffff;
#pragma unroll
    for (int k = 0; k < FPS_P; ++k) {
      const float dx = px[k] - cx;
      const float dy = py[k] - cy;
      const float dz = pz[k] - cz;
      const float d  = __fmaf_rn(dx, dx, __fmaf_rn(dy, dy, dz * dz));
      const float nd = fminf(dist[k], d);
      dist[k] = nd;
      const int p = tid + k * FPS_BLOCK;
      if (nd > bestv || (nd == bestv && p < besti)) { bestv = nd; besti = p; }
    }

    // wave32 butterfly argmax (first-index-wins on ties, like jnp.argmax)
#pragma unroll
    for (int off = 16; off > 0; off >>= 1) {
      const float ov = __shfl_xor(bestv, off, 32);
      const int   oi = __shfl_xor(besti, off, 32);
      if (ov > bestv || (ov == bestv && oi < besti)) { bestv = ov; besti = oi; }
    }
    if (lane == 0) { redv[wave] = bestv; redi[wave] = besti; }
    __syncthreads();

    if (tid < 32) {  // wave 0 reduces the 32 per-wave winners
      float v = redv[lane];
      int   i = redi[lane];
#pragma unroll
      for (int off = 16; off > 0; off >>= 1) {
        const float ov = __shfl_xor(v, off, 32);
        const int   oi = __shfl_xor(i, off, 32);
        if (ov > v || (ov == v && oi < i)) { v = ov; i = oi; }
      }
      if (lane == 0) {
        *swin = i;
        idx_out[(size_t)b * FPS_NS + it] = cur;  // scan emits 'farthest' before update
      }
    }
    __syncthreads();
    cur = *swin;
  }
}

// One wave per sampled row: 128-ch feat row = 32 lanes x float4 (b128 ld/st), 3-ch xyz.
__global__ __launch_bounds__(256) void gather_kernel(const float* __restrict__ xyz,
                                                     const float* __restrict__ feat,
                                                     const int* __restrict__ sidx,
                                                     float* __restrict__ out_xyz,
                                                     float* __restrict__ out_feat) {
  const int gw   = (int)((blockIdx.x * blockDim.x + threadIdx.x) >> 5);
  const int lane = threadIdx.x & 31;
  if (gw >= FPS_B * FPS_NS) return;
  const int b = gw >> 12;            // / FPS_NS
  const int s = gw & (FPS_NS - 1);
  const int p = sidx[(size_t)b * FPS_NS + s];

  const float4* src = (const float4*)(feat + ((size_t)b * FPS_N + p) * FPS_C);
  float4*       dst = (float4*)(out_feat + ((size_t)b * FPS_NS + s) * FPS_C);
  dst[lane] = src[lane];

  if (lane < 3)
    out_xyz[((size_t)b * FPS_NS + s) * 3 + lane] =
        xyz[((size_t)b * FPS_N + p) * 3 + lane];
}

extern "C" void kernel_launch(void* const* d_in, const int* in_sizes, int n_in,
                              void* d_out, int out_size, void* d_ws, size_t ws_size,
                              hipStream_t stream) {
  (void)in_sizes; (void)n_in; (void)out_size; (void)ws_size;
  const float* xyz  = (const float*)d_in[0];
  const float* feat = (const float*)d_in[1];
  float* out_xyz  = (float*)d_out;
  float* out_feat = (float*)d_out + (size_t)FPS_B * FPS_NS * 3;
  int*   sidx     = (int*)d_ws;   // [B, NS] sampled indices

  // Phase 1: one workgroup per batch; xyz slab LDS-resident via TDM.
  fps_kernel<<<FPS_B, FPS_BLOCK, LDS_TOTAL, stream>>>(xyz, sidx);

  // Phase 2: bandwidth-bound gather, 8 rows (waves) per 256-thread block.
  const int rows   = FPS_B * FPS_NS;       // 131072
  const int blocks = rows / 8;             // 16384
  gather_kernel<<<blocks, 256, 0, stream>>>(xyz, feat, sidx, out_xyz, out_feat);
}